// SNN_31009663877469
// MI455X (gfx1250) — compile-verified
//
#include <hip/hip_runtime.h>

// ALIF constants from the reference
#define ALPHA   0.95f
#define RHO     0.9f
#define BETA    1.8f
#define THBASE  1.0f

#define Bsz 16
#define Nn  4096
#define Dd  8
#define NFFk 1024
#define BN  (Bsz * Nn)      // 65536
#define NCHUNK (Dd + 1)     // 8 internal-delay chunks + 1 feedforward chunk

typedef float v2f __attribute__((ext_vector_type(2)));
typedef float v8f __attribute__((ext_vector_type(8)));

// One wave per (16-col output tile, K-chunk). chunk 0..7 -> delay d, chunk 8 -> FF.
// Computes a 16x16 f32 partial tile with V_WMMA_F32_16X16X4_F32 and stores it
// (no atomics -> deterministic).
__global__ void __launch_bounds__(32) snn_gemm_kernel(
    const float* __restrict__ Xd,    // [D, B, N]
    const float* __restrict__ FF,    // [B, NFF]
    const float* __restrict__ Wint,  // [D, N, N]
    const float* __restrict__ Wff,   // [NFF, N]
    float* __restrict__ partial)     // [NCHUNK, B, N]
{
    const int lane  = threadIdx.x;        // 0..31
    const int m0    = blockIdx.x << 4;    // output column tile base
    const int chunk = blockIdx.y;         // 0..8
    const int nl    = lane & 15;          // A: row M / B: col-in-tile (ISA layout)
    const int kb    = (lane >> 4) << 1;   // K base per half-wave: 0 or 2

    v8f c = {0.f, 0.f, 0.f, 0.f, 0.f, 0.f, 0.f, 0.f};

    if (chunk < Dd) {
        // A = Xd[d, b=nl, :], row stride N;  B = Wint[d, k, m0+nl], row stride N
        const float* __restrict__ Ap = Xd + ((size_t)chunk * Bsz + nl) * Nn + kb;
        const float* __restrict__ Bp = Wint + ((size_t)chunk * Nn + kb) * Nn + m0 + nl;
        #pragma unroll 4
        for (int k = 0; k < Nn; k += 4) {
            v2f av = *(const v2f*)(Ap + k);          // K consecutive -> 8B load
            v2f bv;
            bv.x = Bp[(size_t)k * Nn];               // row k+kb
            bv.y = Bp[(size_t)(k + 1) * Nn];         // row k+kb+1
            c = __builtin_amdgcn_wmma_f32_16x16x4_f32(
                    /*neg_a=*/false, av, /*neg_b=*/false, bv,
                    /*c_mod=*/(short)0, c, /*reuse_a=*/false, /*reuse_b=*/false);
        }
    } else {
        // Feedforward: A = FF[b=nl, :], row stride NFF;  B = Wff[k, m0+nl]
        const float* __restrict__ Ap = FF + (size_t)nl * NFFk + kb;
        const float* __restrict__ Bp = Wff + (size_t)kb * Nn + m0 + nl;
        #pragma unroll 4
        for (int k = 0; k < NFFk; k += 4) {
            v2f av = *(const v2f*)(Ap + k);
            v2f bv;
            bv.x = Bp[(size_t)k * Nn];
            bv.y = Bp[(size_t)(k + 1) * Nn];
            c = __builtin_amdgcn_wmma_f32_16x16x4_f32(
                    false, av, false, bv, (short)0, c, false, false);
        }
    }

    // C/D layout: VGPR r -> batch row r + 8*(lane>=16), column m0 + (lane&15)
    float* __restrict__ outp = partial + (size_t)chunk * BN
                             + (size_t)((lane >> 4) << 3) * Nn + m0 + nl;
    #pragma unroll
    for (int r = 0; r < 8; ++r)
        outp[(size_t)r * Nn] = c[r];
}

// Sum the 9 partials (fixed order -> deterministic) and apply the ALIF update.
__global__ void __launch_bounds__(256) snn_epilogue_kernel(
    const float* __restrict__ V,
    const float* __restrict__ A,
    const float* __restrict__ partial,
    float* __restrict__ out)            // [3, B, N]
{
    const int i = blockIdx.x * 256 + threadIdx.x;   // 0..BN-1
    float cur = 0.f;
    #pragma unroll
    for (int cch = 0; cch < NCHUNK; ++cch)
        cur += partial[(size_t)cch * BN + i];

    const float v = V[i];
    const float a = A[i];
    const float x = (v - (THBASE + BETA * a)) > 0.f ? 1.f : 0.f;

    out[i]           = x;                               // X
    out[BN + i]      = ALPHA * v * (1.f - x) + cur;     // V_new
    out[2 * BN + i]  = RHO * a + x;                     // a_new
}

extern "C" void kernel_launch(void* const* d_in, const int* in_sizes, int n_in,
                              void* d_out, int out_size, void* d_ws, size_t ws_size,
                              hipStream_t stream) {
    // setup_inputs order: FF, V, a, Xd, W_int, W_ff  (all float32)
    const float* FF   = (const float*)d_in[0];
    const float* V    = (const float*)d_in[1];
    const float* a    = (const float*)d_in[2];
    const float* Xd   = (const float*)d_in[3];
    const float* Wint = (const float*)d_in[4];
    const float* Wff  = (const float*)d_in[5];
    float* out = (float*)d_out;
    float* partial = (float*)d_ws;      // NCHUNK * BN floats = 2.25 MB

    dim3 gemm_grid(Nn / 16, NCHUNK);    // 256 x 9 waves
    snn_gemm_kernel<<<gemm_grid, 32, 0, stream>>>(Xd, FF, Wint, Wff, partial);
    snn_epilogue_kernel<<<BN / 256, 256, 0, stream>>>(V, a, partial, out);
}